// CKA_66846870995416
// MI455X (gfx1250) — compile-verified
//
#include <hip/hip_runtime.h>
#include <math.h>

typedef __attribute__((ext_vector_type(2))) float v2f;
typedef __attribute__((ext_vector_type(8))) float v8f;

#define N 4096
#define D 768
#define TILE 64
#define KCHUNK 32
#define PITCH 36   // LDS row pitch (floats): 36*l mod 64 = 4*(9l mod 16) -> perfect bank coverage

// ---------------- zero accumulators ----------------
__global__ void cka_zero_kernel(float* p, int n) {
    int i = blockIdx.x * blockDim.x + threadIdx.x;
    if (i < n) p[i] = 0.0f;
}

// ---------------- squared row norms ----------------
__global__ __launch_bounds__(256)
void cka_sqnorm_kernel(const float* __restrict__ X, const float* __restrict__ Y,
                       float* __restrict__ dx, float* __restrict__ dy) {
    int wave = threadIdx.x >> 5;
    int lane = threadIdx.x & 31;
    int row  = blockIdx.x * 8 + wave;
    const float* xr = X + (size_t)row * D;
    const float* yr = Y + (size_t)row * D;
    float sx = 0.0f, sy = 0.0f;
    for (int k = lane; k < D; k += 32) {
        float a = xr[k]; sx += a * a;
        float b = yr[k]; sy += b * b;
    }
    for (int m = 16; m >= 1; m >>= 1) {
        sx += __shfl_xor(sx, m, 32);
        sy += __shfl_xor(sy, m, 32);
    }
    if (lane == 0) { dx[row] = sx; dy[row] = sy; }
}

// ---------------- main fused tile kernel ----------------
__global__ __launch_bounds__(256)
void cka_tile_kernel(const float* __restrict__ X, const float* __restrict__ Y,
                     const float* __restrict__ dx, const float* __restrict__ dy,
                     float* __restrict__ rsx, float* __restrict__ rsy,
                     float* __restrict__ s3, const int* __restrict__ sigp) {
    __shared__ __align__(16) float ldsXi[TILE * PITCH];
    __shared__ __align__(16) float ldsXj[TILE * PITCH];
    __shared__ __align__(16) float ldsYi[TILE * PITCH];
    __shared__ __align__(16) float ldsYj[TILE * PITCH];
    __shared__ float rs_x[TILE];
    __shared__ float rs_y[TILE];
    __shared__ float blk3[3];

    const int tid  = threadIdx.x;
    const int lane = tid & 31;
    const int wave = tid >> 5;
    const int i0   = blockIdx.y * TILE;
    const int j0   = blockIdx.x * TILE;

    if (tid < TILE) { rs_x[tid] = 0.0f; rs_y[tid] = 0.0f; }
    if (tid < 3)    blk3[tid] = 0.0f;

    // wave -> subtile mapping: column subtile sc, row subtiles sr0 and sr0+2
    const int sc  = wave & 3;
    const int sr0 = wave >> 2;        // 0 or 1
    const int sr1 = sr0 + 2;

    const int half = lane >> 4;       // 0/1 half-wave
    const int l15  = lane & 15;

    // cooperative load mapping: 256 threads x 8 floats = 64x32 chunk
    const int lr = tid >> 2;          // row within tile 0..63
    const int lc = (tid & 3) * 8;     // col group start {0,8,16,24}

    v8f accx0 = {}, accx1 = {}, accy0 = {}, accy1 = {};

    for (int k0 = 0; k0 < D; k0 += KCHUNK) {
        const float4* gxi = (const float4*)(X + (size_t)(i0 + lr) * D + k0 + lc);
        const float4* gxj = (const float4*)(X + (size_t)(j0 + lr) * D + k0 + lc);
        const float4* gyi = (const float4*)(Y + (size_t)(i0 + lr) * D + k0 + lc);
        const float4* gyj = (const float4*)(Y + (size_t)(j0 + lr) * D + k0 + lc);
        float4 a0 = gxi[0], a1 = gxi[1];
        float4 b0 = gxj[0], b1 = gxj[1];
        float4 c0 = gyi[0], c1 = gyi[1];
        float4 e0 = gyj[0], e1 = gyj[1];

        __syncthreads();  // previous chunk's compute done (also covers rs_x init)
        *(float4*)(ldsXi + lr * PITCH + lc)     = a0;
        *(float4*)(ldsXi + lr * PITCH + lc + 4) = a1;
        *(float4*)(ldsXj + lr * PITCH + lc)     = b0;
        *(float4*)(ldsXj + lr * PITCH + lc + 4) = b1;
        *(float4*)(ldsYi + lr * PITCH + lc)     = c0;
        *(float4*)(ldsYi + lr * PITCH + lc + 4) = c1;
        *(float4*)(ldsYj + lr * PITCH + lc)     = e0;
        *(float4*)(ldsYj + lr * PITCH + lc + 4) = e1;
        __syncthreads();

#pragma unroll
        for (int ks = 0; ks < KCHUNK / 4; ++ks) {
            const int cc = ks * 4 + half * 2;
            v2f bx  = *(const v2f*)(ldsXj + (sc  * 16 + l15) * PITCH + cc);
            v2f by  = *(const v2f*)(ldsYj + (sc  * 16 + l15) * PITCH + cc);
            v2f ax0 = *(const v2f*)(ldsXi + (sr0 * 16 + l15) * PITCH + cc);
            v2f ax1 = *(const v2f*)(ldsXi + (sr1 * 16 + l15) * PITCH + cc);
            v2f ay0 = *(const v2f*)(ldsYi + (sr0 * 16 + l15) * PITCH + cc);
            v2f ay1 = *(const v2f*)(ldsYi + (sr1 * 16 + l15) * PITCH + cc);
            accx0 = __builtin_amdgcn_wmma_f32_16x16x4_f32(false, ax0, false, bx, (short)0, accx0, false, false);
            accx1 = __builtin_amdgcn_wmma_f32_16x16x4_f32(false, ax1, false, bx, (short)0, accx1, false, false);
            accy0 = __builtin_amdgcn_wmma_f32_16x16x4_f32(false, ay0, false, by, (short)0, accy0, false, false);
            accy1 = __builtin_amdgcn_wmma_f32_16x16x4_f32(false, ay1, false, by, (short)0, accy1, false, false);
        }
    }

    // ---- turn Gram accumulators into RBF kernel elements + statistics ----
    float sg   = (float)sigp[0];
    float invs = 0.5f / (sg * sg);           // exp((2G - di - dj) * invs)

    float dxj = dx[j0 + sc * 16 + l15];
    float dyj = dy[j0 + sc * 16 + l15];

    float sxy = 0.0f, sxx = 0.0f, syy = 0.0f;

#pragma unroll
    for (int sub = 0; sub < 2; ++sub) {
        const int sr   = sub ? sr1 : sr0;
        const v8f  gx  = sub ? accx1 : accx0;
        const v8f  gy  = sub ? accy1 : accy0;
        const int rbase = sr * 16 + half * 8;      // row within 64-tile of element v
#pragma unroll
        for (int v = 0; v < 8; ++v) {
            float dxi = dx[i0 + rbase + v];
            float dyi = dy[i0 + rbase + v];
            float kx = __expf((2.0f * gx[v] - dxi - dxj) * invs);
            float ky = __expf((2.0f * gy[v] - dyi - dyj) * invs);
            sxy += kx * ky;
            sxx += kx * kx;
            syy += ky * ky;
            // 16-lane row sum (masks <16 stay within each half-wave)
            float rx = kx, ry = ky;
            rx += __shfl_xor(rx, 1); ry += __shfl_xor(ry, 1);
            rx += __shfl_xor(rx, 2); ry += __shfl_xor(ry, 2);
            rx += __shfl_xor(rx, 4); ry += __shfl_xor(ry, 4);
            rx += __shfl_xor(rx, 8); ry += __shfl_xor(ry, 8);
            if (l15 == 0) {
                atomicAdd(&rs_x[rbase + v], rx);
                atomicAdd(&rs_y[rbase + v], ry);
            }
        }
    }

    // full-wave reduce scalar sums
    for (int m = 1; m < 32; m <<= 1) {
        sxy += __shfl_xor(sxy, m);
        sxx += __shfl_xor(sxx, m);
        syy += __shfl_xor(syy, m);
    }
    if (lane == 0) {
        atomicAdd(&blk3[0], sxy);
        atomicAdd(&blk3[1], sxx);
        atomicAdd(&blk3[2], syy);
    }
    __syncthreads();

    if (tid < TILE) {
        atomicAdd(&rsx[i0 + tid], rs_x[tid]);
        atomicAdd(&rsy[i0 + tid], rs_y[tid]);
    }
    if (tid < 3) atomicAdd(&s3[tid], blk3[tid]);
}

// ---------------- finalize ----------------
__global__ __launch_bounds__(256)
void cka_final_kernel(const float* __restrict__ rsx, const float* __restrict__ rsy,
                      const float* __restrict__ s3, float* __restrict__ out) {
    __shared__ float sm[8][5];
    int lane = threadIdx.x & 31, wave = threadIdx.x >> 5;
    float tx = 0, ty = 0, pxy = 0, pxx = 0, pyy = 0;
    for (int i = threadIdx.x; i < N; i += 256) {
        float a = rsx[i], b = rsy[i];
        tx += a; ty += b; pxy += a * b; pxx += a * a; pyy += b * b;
    }
    for (int m = 1; m < 32; m <<= 1) {
        tx  += __shfl_xor(tx, m);  ty  += __shfl_xor(ty, m);
        pxy += __shfl_xor(pxy, m); pxx += __shfl_xor(pxx, m); pyy += __shfl_xor(pyy, m);
    }
    if (lane == 0) { sm[wave][0]=tx; sm[wave][1]=ty; sm[wave][2]=pxy; sm[wave][3]=pxx; sm[wave][4]=pyy; }
    __syncthreads();
    if (threadIdx.x == 0) {
        float Tx=0, Ty=0, Dxy=0, Dxx=0, Dyy=0;
        for (int w = 0; w < 8; ++w) {
            Tx += sm[w][0]; Ty += sm[w][1]; Dxy += sm[w][2]; Dxx += sm[w][3]; Dyy += sm[w][4];
        }
        const float n  = (float)N;
        const float n2 = n * n;
        float hxy = s3[0] - (2.0f / n) * Dxy + (Tx * Ty) / n2;
        float hxx = s3[1] - (2.0f / n) * Dxx + (Tx * Tx) / n2;
        float hyy = s3[2] - (2.0f / n) * Dyy + (Ty * Ty) / n2;
        out[0] = hxy / (sqrtf(hxx) * sqrtf(hyy));
    }
}

extern "C" void kernel_launch(void* const* d_in, const int* in_sizes, int n_in,
                              void* d_out, int out_size, void* d_ws, size_t ws_size,
                              hipStream_t stream) {
    const float* X   = (const float*)d_in[0];
    const float* Y   = (const float*)d_in[1];
    const int*   sig = (const int*)d_in[2];

    float* ws  = (float*)d_ws;
    float* s3  = ws;                 // 3 scalars (+pad to 16)
    float* rsx = ws + 16;            // N
    float* rsy = ws + 16 + N;        // N
    float* dxs = ws + 16 + 2 * N;    // N
    float* dys = ws + 16 + 3 * N;    // N

    int nz = 16 + 2 * N;
    cka_zero_kernel<<<(nz + 255) / 256, 256, 0, stream>>>(ws, nz);
    cka_sqnorm_kernel<<<N / 8, 256, 0, stream>>>(X, Y, dxs, dys);

    dim3 grid(N / TILE, N / TILE);
    cka_tile_kernel<<<grid, 256, 0, stream>>>(X, Y, dxs, dys, rsx, rsy, s3, sig);

    cka_final_kernel<<<1, 256, 0, stream>>>(rsx, rsy, s3, (float*)d_out);
}